// Quantumnet_670014898507
// MI455X (gfx1250) — compile-verified
//
#include <hip/hip_runtime.h>

typedef float v2f __attribute__((ext_vector_type(2)));
typedef float v8f __attribute__((ext_vector_type(8)));

constexpr int NQ     = 4;
constexpr int DEPTH  = 6;
constexpr int DIN    = 512;
constexpr int BLOCK  = 256;
constexpr int KGROUPS = DIN / 4;   // 128 k-steps of 4

// CNOT(control bit CB, target bit TB): swap amplitudes where control=1.
template<int CB, int TB>
__device__ __forceinline__ void cnot_g(float* s) {
#pragma unroll
  for (int i = 0; i < 16; ++i) {
    if ((i & CB) && !(i & TB)) {
      float t = s[i]; s[i] = s[i | TB]; s[i | TB] = t;
    }
  }
}

// RY on wire with bit BIT: [c -s; s c] (half-angle c/s passed in).
template<int BIT>
__device__ __forceinline__ void ry_g(float* s, float ct, float st) {
#pragma unroll
  for (int i = 0; i < 16; ++i) {
    if (!(i & BIT)) {
      float a = s[i], b = s[i | BIT];
      s[i]       = ct * a - st * b;
      s[i | BIT] = st * a + ct * b;
    }
  }
}

__global__ __launch_bounds__(BLOCK) void qnet_fused(
    const float* __restrict__ x,       // (B, 512)
    const float* __restrict__ pre_w,   // (4, 512)
    const float* __restrict__ pre_b,   // (4,)
    const float* __restrict__ qp,      // (24,)
    const float* __restrict__ post_w,  // (2, 4)
    const float* __restrict__ post_b,  // (2,)
    float* __restrict__ out)           // (B, 2)
{
  // B operand pre-staged in exact WMMA fragment layout:
  // entry e = g*32 + lane holds B[k=4g+2*(lane>>4)+{0,1}][n=lane&15] (0 for n>=4).
  __shared__ v2f   lds_bfrag[KGROUPS * 32];   // 32 KB, conflict-free stream
  __shared__ float lds_q[BLOCK * NQ];         // 4 KB: pre-net raw outputs
  __shared__ float lds_cs[DEPTH * NQ * 2];    // uniform q_param cos/sin pairs

  const int tid  = threadIdx.x;
  const int lane = tid & 31;
  const int wave = tid >> 5;
  const int m = lane & 15;   // A: row-in-tile. B/D: N column.
  const int h = lane >> 4;   // lane half -> K offset 2*h (ISA f32 A/B layout)

  // Stage B fragments (zero-padded N columns 4..15), 16 entries per thread.
  for (int e = tid; e < KGROUPS * 32; e += BLOCK) {
    int l  = e & 31;
    int n  = l & 15;
    int k  = 4 * (e >> 5) + 2 * (l >> 4);
    v2f val = {0.f, 0.f};
    if (n < NQ) { val[0] = pre_w[n * DIN + k]; val[1] = pre_w[n * DIN + k + 1]; }
    lds_bfrag[e] = val;
  }
  if (tid < DEPTH * NQ) {
    float sv, cv;
    __sincosf(qp[tid] * 0.5f, &sv, &cv);
    lds_cs[tid * 2]     = cv;
    lds_cs[tid * 2 + 1] = sv;
  }
  __syncthreads();

  // ---------- Phase 1: pre_net GEMM via V_WMMA_F32_16X16X4_F32 ----------
  // Each wave: 2 tiles of 16 samples, 4 independent accumulator chains.
  const size_t rowBase = (size_t)blockIdx.x * BLOCK + (size_t)wave * 32;
  const float* a0 = x + (rowBase + (size_t)m) * DIN + 2 * h;
  const float* a1 = a0 + (size_t)16 * DIN;
  const v2f*   bptr = &lds_bfrag[lane];

  v8f c00 = {0.f,0.f,0.f,0.f,0.f,0.f,0.f,0.f};
  v8f c01 = c00, c10 = c00, c11 = c00;
#pragma unroll 2
  for (int g = 0; g < KGROUPS; g += 2) {
    v2f av0a = __builtin_nontemporal_load((const v2f*)(a0 + 4 * g));
    v2f av1a = __builtin_nontemporal_load((const v2f*)(a1 + 4 * g));
    v2f av0b = __builtin_nontemporal_load((const v2f*)(a0 + 4 * g + 4));
    v2f av1b = __builtin_nontemporal_load((const v2f*)(a1 + 4 * g + 4));
    v2f bva  = bptr[g * 32];
    v2f bvb  = bptr[g * 32 + 32];
    // 8 args: (neg_a, A, neg_b, B, c_mod, C, reuse_a, reuse_b)
    c00 = __builtin_amdgcn_wmma_f32_16x16x4_f32(false, av0a, false, bva,
                                                (short)0, c00, false, false);
    c10 = __builtin_amdgcn_wmma_f32_16x16x4_f32(false, av1a, false, bva,
                                                (short)0, c10, false, false);
    c01 = __builtin_amdgcn_wmma_f32_16x16x4_f32(false, av0b, false, bvb,
                                                (short)0, c01, false, false);
    c11 = __builtin_amdgcn_wmma_f32_16x16x4_f32(false, av1b, false, bvb,
                                                (short)0, c11, false, false);
  }
  v8f c0 = c00 + c01;
  v8f c1 = c10 + c11;

  // D layout: VGPR v holds M=v (lanes 0-15) / M=v+8 (lanes 16-31), N=lane%16.
  if (m < NQ) {
    float bias = pre_b[m];
#pragma unroll
    for (int v = 0; v < 8; ++v) {
      int sit = v + 8 * h; // sample index within 16-row tile
      lds_q[(wave * 32 + sit) * NQ + m]      = c0[v] + bias;
      lds_q[(wave * 32 + 16 + sit) * NQ + m] = c1[v] + bias;
    }
  }
  __syncthreads();

  // ---------- Phase 2: 4-qubit statevector sim, one sample per thread ----
  float s[16];
#pragma unroll
  for (int i = 0; i < 16; ++i) s[i] = 0.25f; // H^4 |0000> = uniform 1/4

  // Per-sample encoding RY(tanh(q)*pi/2)
#pragma unroll
  for (int w = 0; w < NQ; ++w) {
    float ang = tanhf(lds_q[tid * NQ + w]) * 1.5707963267948966f;
    float sv, cv;
    __sincosf(ang * 0.5f, &sv, &cv);
    if (w == 0) ry_g<8>(s, cv, sv);
    if (w == 1) ry_g<4>(s, cv, sv);
    if (w == 2) ry_g<2>(s, cv, sv);
    if (w == 3) ry_g<1>(s, cv, sv);
  }

#pragma unroll
  for (int k = 0; k < DEPTH; ++k) {
    cnot_g<8, 4>(s);  // CNOT(0,1)
    cnot_g<2, 1>(s);  // CNOT(2,3)
    cnot_g<4, 2>(s);  // CNOT(1,2)
#pragma unroll
    for (int w = 0; w < NQ; ++w) {
      float cv = lds_cs[(k * NQ + w) * 2];
      float sv = lds_cs[(k * NQ + w) * 2 + 1];
      if (w == 0) ry_g<8>(s, cv, sv);
      if (w == 1) ry_g<4>(s, cv, sv);
      if (w == 2) ry_g<2>(s, cv, sv);
      if (w == 3) ry_g<1>(s, cv, sv);
    }
  }

  // <Z_w> = sum (+/- |amp|^2)
  float z0 = 0.f, z1 = 0.f, z2 = 0.f, z3 = 0.f;
#pragma unroll
  for (int i = 0; i < 16; ++i) {
    float p = s[i] * s[i];
    z0 += (i & 8) ? -p : p;
    z1 += (i & 4) ? -p : p;
    z2 += (i & 2) ? -p : p;
    z3 += (i & 1) ? -p : p;
  }

  // ---------- Phase 3: post_net (4 -> 2) ----------
  float o0 = post_b[0] + z0 * post_w[0] + z1 * post_w[1] + z2 * post_w[2] + z3 * post_w[3];
  float o1 = post_b[1] + z0 * post_w[4] + z1 * post_w[5] + z2 * post_w[6] + z3 * post_w[7];

  size_t sidx = (size_t)blockIdx.x * BLOCK + tid;
  v2f ov = {o0, o1};
  *(v2f*)(out + sidx * 2) = ov;
}

extern "C" void kernel_launch(void* const* d_in, const int* in_sizes, int n_in,
                              void* d_out, int out_size, void* d_ws, size_t ws_size,
                              hipStream_t stream) {
  (void)n_in; (void)out_size; (void)d_ws; (void)ws_size;
  const float* x      = (const float*)d_in[0];
  const float* pre_w  = (const float*)d_in[1];
  const float* pre_b  = (const float*)d_in[2];
  const float* qp     = (const float*)d_in[3];
  const float* post_w = (const float*)d_in[4];
  const float* post_b = (const float*)d_in[5];
  float* out = (float*)d_out;

  const int B    = in_sizes[0] / DIN;   // 262144
  const int grid = B / BLOCK;           // 1024 blocks of 256 threads (8 waves)
  qnet_fused<<<grid, BLOCK, 0, stream>>>(x, pre_w, pre_b, qp, post_w, post_b, out);
}